// glimpse_module_60284160967423
// MI455X (gfx1250) — compile-verified
//
#include <hip/hip_runtime.h>

// retina glimpse: B=64, C=3, H=W=512, G=64, K=3 scales (64,128,256), S=2
// out: (B, K, C, 64, 64) float32, flat idx = ((b*3 + k)*3 + c)*4096 + gy*64 + gx

__device__ __forceinline__ void get_coords(const float* __restrict__ l, int b,
                                           int& cy, int& cx) {
    // match reference: int32(0.5f * ((l + 1) * 512)), values are >= 0 so trunc==floor
    float l0 = l[2 * b + 0];
    float l1 = l[2 * b + 1];
    cy = (int)(0.5f * ((l0 + 1.0f) * 512.0f));
    cx = (int)(0.5f * ((l1 + 1.0f) * 512.0f));
}

// Scales k=0 (64x64, f=1) and k=1 (128x128 -> avgpool2 -> 64x64).
// grid = (32, 3, 64): x = k*16 + tile (tile = 256-output chunk of the 64x64 map),
// y = c, z = b. b,c uniform per block -> coords load scalarizes to s_load.
__global__ void retina_small(const float* __restrict__ x,
                             const float* __restrict__ l,
                             float* __restrict__ out) {
    int k    = (int)blockIdx.x >> 4;         // 0 or 1
    int tile = (int)blockIdx.x & 15;
    int c    = (int)blockIdx.y;
    int b    = (int)blockIdx.z;
    int p    = tile * 256 + (int)threadIdx.x;  // 0..4095 within the 64x64 map
    int gx   = p & 63;
    int gy   = p >> 6;

    int cy, cx;
    get_coords(l, b, cy, cx);
    const float* __restrict__ xp = x + (((size_t)(b * 3 + c)) << 18); // *512*512

    float v;
    if (k == 0) {
        int r = cy - 32 + gy;
        int q = cx - 32 + gx;
        v = (r >= 0 && r < 512 && q >= 0 && q < 512) ? xp[(r << 9) + q] : 0.0f;
    } else {
        int r0 = cy - 64 + 2 * gy;
        int q0 = cx - 64 + 2 * gx;
        float s = 0.0f;
#pragma unroll
        for (int dy = 0; dy < 2; ++dy) {
            int r = r0 + dy;
            if (r >= 0 && r < 512) {
#pragma unroll
                for (int dx = 0; dx < 2; ++dx) {
                    int q = q0 + dx;
                    if (q >= 0 && q < 512) s += xp[(r << 9) + q];
                }
            }
        }
        v = s * 0.25f;
    }
    out[((size_t)(b * 9 + k * 3 + c) << 12) + (gy << 6) + gx] = v;
}

// Scale k=2 (256x256 -> avgpool4 -> 64x64).
// grid = (16, 3, 64): x = group-of-4 output rows, y = c, z = b.
// Stage 16 input rows x 256 cols through LDS with gfx1250 async global->LDS
// copies (ASYNCcnt); per-lane EXEC predication handles zero padding (OOB slots
// ds_store'd to 0; each LDS slot has exactly one writer). Pool via b128 LDS reads.
__global__ void retina_large(const float* __restrict__ x,
                             const float* __restrict__ l,
                             float* __restrict__ out) {
    __shared__ float tile[16 * 256];         // 16 KB

    int gy4 = (int)blockIdx.x;               // 0..15
    int c   = (int)blockIdx.y;
    int b   = (int)blockIdx.z;
    int t   = (int)threadIdx.x;              // 0..255

    int cy, cx;
    get_coords(l, b, cy, cx);
    const float* __restrict__ xp = x + (((size_t)(b * 3 + c)) << 18);

    int  gcol  = cx - 128 + t;               // this thread's input column
    bool colok = (gcol >= 0) && (gcol < 512);
    int  row0  = cy - 128 + gy4 * 16;        // uniform per block

#pragma unroll
    for (int i = 0; i < 16; ++i) {           // 16 input rows for 4 output rows
        int grow = row0 + i;
        int e    = i * 256 + t;              // exclusive LDS slot for this thread
        if (colok && grow >= 0 && grow < 512) {
            const float* gp = xp + ((size_t)grow << 9) + gcol;
            // LDS byte address = low 32 bits of the flat shared pointer
            unsigned ldsa = (unsigned)(unsigned long long)(&tile[e]);
            asm volatile("global_load_async_to_lds_b32 %0, %1, off"
                         :: "v"(ldsa), "v"(gp)
                         : "memory");
        } else {
            tile[e] = 0.0f;                  // zero padding (DScnt path)
        }
    }

    // Wait for this wave's async transfers to land in LDS, then cross-wave sync
    // (__syncthreads also waits DScnt for the zero stores).
#if defined(__has_builtin)
#if __has_builtin(__builtin_amdgcn_s_wait_asynccnt)
    __builtin_amdgcn_s_wait_asynccnt(0);
#else
    asm volatile("s_wait_asynccnt 0" ::: "memory");
#endif
#else
    asm volatile("s_wait_asynccnt 0" ::: "memory");
#endif
    __syncthreads();

    int ry = t >> 6;                         // 0..3 -> output row within group
    int gx = t & 63;
    // 4x4 window, rows via 16B-aligned float4 LDS loads (ds_load_b128)
    float s = 0.0f;
#pragma unroll
    for (int dy = 0; dy < 4; ++dy) {
        const float4* rowp =
            reinterpret_cast<const float4*>(&tile[(ry * 4 + dy) * 256]);
        float4 w = rowp[gx];
        s += (w.x + w.y) + (w.z + w.w);
    }
    int gy = gy4 * 4 + ry;
    out[((size_t)(b * 9 + 6 + c) << 12) + (gy << 6) + gx] = s * (1.0f / 16.0f);
}

extern "C" void kernel_launch(void* const* d_in, const int* in_sizes, int n_in,
                              void* d_out, int out_size, void* d_ws, size_t ws_size,
                              hipStream_t stream) {
    (void)in_sizes; (void)n_in; (void)out_size; (void)d_ws; (void)ws_size;
    const float* x = (const float*)d_in[0];   // (64,3,512,512)
    const float* l = (const float*)d_in[1];   // (64,2)
    float* out = (float*)d_out;               // (64,3,3,64,64)

    // scales 0+1: k in {0,1} x 16 tiles, c, b
    retina_small<<<dim3(32, 3, 64), 256, 0, stream>>>(x, l, out);
    // scale 2: 16 row-groups, c, b; LDS-staged async copies
    retina_large<<<dim3(16, 3, 64), 256, 0, stream>>>(x, l, out);
}